// DepthWiseConv1d_12240656793682
// MI455X (gfx1250) — compile-verified
//
#include <hip/hip_runtime.h>

typedef float v4f __attribute__((ext_vector_type(4)));
typedef __attribute__((address_space(3))) v4f lds_v4f;  // raw LDS (ds-offset) space

#define B_       8
#define L_       4096
#define C_       2048
#define LOUT_    4099                     // L + 2*PAD - K + 1
#define C4_      512                      // C/4 float4 groups per row
#define CHUNK_   64                       // output rows per thread
#define NCHUNK_  65                       // ceil(LOUT_/CHUNK_)
#define RING_    16                       // async LDS ring depth (rows)
#define THREADS_ 256

// out[b,l,c] = bias[c] + sum_{k=0..3} x[b, l+k-3, c] * w[c,k]  (x rows OOB -> 0)
//
// x rows stream HBM -> LDS via 16-deep ASYNCcnt pipeline (global_load_async_to_lds_b128),
// compute reads LDS (conflict-free [slot][lane] layout), NT b128 stores to out.
__global__ __launch_bounds__(THREADS_) void dwconv1d_async_kernel(
    const float* __restrict__ x,      // (B, L, C)
    const float* __restrict__ w,      // (C, K=4): per-channel taps = one float4
    const float* __restrict__ bias,   // (C,)
    float* __restrict__ out)          // (B, LOUT, C)
{
    // [slot][lane]: lane i reads/writes bytes [i*16, i*16+16) of a 4 KB slot row
    // -> fully coalesced LDS banks, no conflicts.
    __shared__ v4f ring[RING_][THREADS_];   // 64 KB per block

    const int tid   = blockIdx.x * THREADS_ + threadIdx.x;
    const int c4    = tid & (C4_ - 1);          // fastest across lanes (coalesced)
    const int rest  = tid >> 9;                 // wave-uniform
    const int chunk = rest % NCHUNK_;
    const int b     = rest / NCHUNK_;
    if (b >= B_) return;                        // never taken (exact grid), defensive

    const v4f* __restrict__ xb = (const v4f*)x + (size_t)b * L_ * C4_ + c4;
    v4f* __restrict__ ob       = (v4f*)out + (size_t)b * LOUT_ * C4_ + c4;

    // Per-channel taps (contiguous float4 each), transposed to per-tap vectors.
    const v4f* wq  = (const v4f*)w;
    const v4f wA   = wq[4 * c4 + 0];
    const v4f wB   = wq[4 * c4 + 1];
    const v4f wC   = wq[4 * c4 + 2];
    const v4f wD   = wq[4 * c4 + 3];
    const v4f wt0  = {wA[0], wB[0], wC[0], wD[0]};
    const v4f wt1  = {wA[1], wB[1], wC[1], wD[1]};
    const v4f wt2  = {wA[2], wB[2], wC[2], wD[2]};
    const v4f wt3  = {wA[3], wB[3], wC[3], wD[3]};
    const v4f bv   = ((const v4f*)bias)[c4];

    const int l0   = chunk * CHUNK_;
    const int lend = (l0 + CHUNK_ < LOUT_) ? (l0 + CHUNK_) : LOUT_;

    // Raw LDS byte offset of this lane's column in slot 0.
    const unsigned lds_base = (unsigned)(size_t)(lds_v4f*)&ring[0][threadIdx.x];

    // Issue one async row load: x row (clamped) -> ring slot ((r+3) & 15).
    // OOB rows load clamped (garbage) data; masked to zero at read time.
    auto issue = [&](int r) {
        const int rc = r < 0 ? 0 : (r >= L_ ? L_ - 1 : r);
        const v4f* ga = xb + (size_t)rc * C4_;
        const unsigned la =
            lds_base + (unsigned)(((r + 3) & (RING_ - 1)) * (THREADS_ * 16));
        asm volatile("global_load_async_to_lds_b128 %0, %1, off th:TH_LOAD_NT"
                     :: "v"(la), "v"(ga)
                     : "memory");
    };

    const v4f zero = {0.f, 0.f, 0.f, 0.f};
    auto rdrow = [&](int r) -> v4f {   // ds_load_b128 with scalar OOB mask
        return ((unsigned)r < (unsigned)L_)
                   ? ring[(r + 3) & (RING_ - 1)][threadIdx.x]
                   : zero;
    };

    // ---- prologue: fill the 16-deep pipeline (rows l0-3 .. l0+12) ----
#pragma unroll
    for (int k = 0; k < RING_; ++k) issue(l0 - 3 + k);

    // oldest 3 done (<=13 outstanding): rows l0-3 .. l0-1 available
    asm volatile("s_wait_asynccnt 0xd" ::: "memory");
    v4f p0 = rdrow(l0 - 3);
    v4f p1 = rdrow(l0 - 2);
    v4f p2 = rdrow(l0 - 1);

    for (int l = l0; l < lend; l += 4) {
        // <=9 outstanding: rows l .. l+3 have landed in LDS (in-order completion)
        asm volatile("s_wait_asynccnt 0x9" ::: "memory");
        v4f n0 = rdrow(l);
        v4f n1 = rdrow(l + 1);
        v4f n2 = rdrow(l + 2);
        v4f n3 = rdrow(l + 3);

        v4f o0 = bv + p0 * wt0 + p1 * wt1 + p2 * wt2 + n0 * wt3;
        v4f o1 = bv + p1 * wt0 + p2 * wt1 + n0 * wt2 + n1 * wt3;
        v4f o2 = bv + p2 * wt0 + n0 * wt1 + n1 * wt2 + n2 * wt3;
        v4f o3 = bv + n0 * wt0 + n1 * wt1 + n2 * wt2 + n3 * wt3;

        __builtin_nontemporal_store(o0, &ob[(size_t)l * C4_]);
        if (l + 1 < lend) __builtin_nontemporal_store(o1, &ob[(size_t)(l + 1) * C4_]);
        if (l + 2 < lend) __builtin_nontemporal_store(o2, &ob[(size_t)(l + 2) * C4_]);
        if (l + 3 < lend) __builtin_nontemporal_store(o3, &ob[(size_t)(l + 3) * C4_]);

        // refill pipeline: rows l+13 .. l+16 reuse slots of rows l-3 .. l
        // (whose ds_load results were already consumed above)
        issue(l + 13);
        issue(l + 14);
        issue(l + 15);
        issue(l + 16);

        p0 = n1; p1 = n2; p2 = n3;
    }
    // leftover async ops drain at s_endpgm (implicit wait-idle)
}

extern "C" void kernel_launch(void* const* d_in, const int* in_sizes, int n_in,
                              void* d_out, int out_size, void* d_ws, size_t ws_size,
                              hipStream_t stream) {
    const float* x    = (const float*)d_in[0];
    const float* w    = (const float*)d_in[1];
    const float* bias = (const float*)d_in[2];
    float* out        = (float*)d_out;

    const int total  = C4_ * NCHUNK_ * B_;        // 512 * 65 * 8 = 266240 threads
    const int blocks = (total + THREADS_ - 1) / THREADS_;  // 1040 blocks
    dwconv1d_async_kernel<<<blocks, THREADS_, 0, stream>>>(x, w, bias, out);
}